// ConvBertLightConv_24558622999247
// MI455X (gfx1250) — compile-verified
//
#include <hip/hip_runtime.h>

// ConvBERT LightConv: dynamic depthwise conv over sequence, fp32 end-to-end,
// mapped onto V_WMMA_F32_16X16X4_F32 (banded 16x24 filter matrix x 24x16 x-slice).

constexpr int kB   = 8;
constexpr int kS   = 4096;
constexpr int kH   = 12;
constexpr int kHD  = 64;
constexpr int kD   = kH * kHD;       // 768
constexpr int kK   = 9;
constexpr int kPAD = kK / 2;         // 4
constexpr int kSBLK = 128;           // seq positions per workgroup
constexpr int kTS   = 16;            // seq positions per wave
constexpr int kROWS = kSBLK + kK - 1;      // 136 staged x rows
constexpr int kRSTR = 72;            // LDS row stride (floats); 72%64=8 -> half-waves hit disjoint banks
constexpr int kFSTR = 12;            // padded filter row stride

typedef __attribute__((ext_vector_type(2))) float v2f;
typedef __attribute__((ext_vector_type(4))) float v4f;
typedef __attribute__((ext_vector_type(8))) float v8f;

__global__ __launch_bounds__(256)
void ConvBertLightConv_wmma_kernel(const float* __restrict__ x,
                                   const float* __restrict__ filt,
                                   float* __restrict__ out)
{
    __shared__ float xt[kROWS * kRSTR];        // x tile (one head), padded rows
    __shared__ float fsm[kSBLK * kFSTR];       // softmax'd filters per position

    const int wg    = blockIdx.x;
    const int nSblk = kS / kSBLK;              // 32
    const int sblk  = wg % nSblk;
    const int h     = (wg / nSblk) % kH;
    const int b     = wg / (nSblk * kH);
    const int sBase = sblk * kSBLK;
    const int tid   = threadIdx.x;

    // ---- Stage x rows [sBase-4, sBase+131] of head h into LDS (float4, coalesced) ----
    const float* xHead = x + ((size_t)b * kS) * kD + (size_t)h * kHD;
    for (int t = tid; t < kROWS * (kHD / 4); t += 256) {
        const int r  = t >> 4;                 // staged row 0..135
        const int c4 = t & 15;                 // float4 column 0..15
        const int s  = sBase - kPAD + r;
        v4f v = {0.f, 0.f, 0.f, 0.f};
        if (s >= 0 && s < kS)
            v = *(const v4f*)(xHead + (size_t)s * kD + c4 * 4);
        *(v4f*)(&xt[r * kRSTR + c4 * 4]) = v;
    }

    // ---- Softmax the 9 taps for each of the 128 positions ----
    if (tid < kSBLK) {
        const int s = sBase + tid;
        const float* fp = filt + ((size_t)b * kS + s) * (kH * kK) + h * kK;
        float f[kK];
        float mx = -1e30f;
#pragma unroll
        for (int k = 0; k < kK; ++k) { f[k] = fp[k]; mx = fmaxf(mx, f[k]); }
        float sum = 0.f;
#pragma unroll
        for (int k = 0; k < kK; ++k) { f[k] = __expf(f[k] - mx); sum += f[k]; }
        const float inv = 1.0f / sum;
#pragma unroll
        for (int k = 0; k < kK; ++k) fsm[tid * kFSTR + k] = f[k] * inv;
    }

    __syncthreads();

    // ---- Per-wave 16(pos) x 64(dim) tile via six K=4 f32 WMMAs per 16-col slab ----
    const int wave = tid >> 5;
    const int lane = tid & 31;
    const int mrow = lane & 15;                // A: M row / B: N col
    const int half = lane >> 4;                // K half selector
    const int srow = wave * kTS;               // wave's first local position

    // Banded A fragments: A[m][j] = fsm[m][j-m], j = 4c + 2*half + {0,1}
    v2f afrag[6];
#pragma unroll
    for (int c = 0; c < 6; ++c) {
        const int j0 = 4 * c + 2 * half;
        const int d0 = j0 - mrow;
        const int d1 = j0 + 1 - mrow;
        float a0 = 0.f, a1 = 0.f;
        if (d0 >= 0 && d0 < kK) a0 = fsm[(srow + mrow) * kFSTR + d0];
        if (d1 >= 0 && d1 < kK) a1 = fsm[(srow + mrow) * kFSTR + d1];
        afrag[c][0] = a0;
        afrag[c][1] = a1;
    }

    float* oBase = out + ((size_t)b * kS + sBase + srow) * kD + (size_t)h * kHD;

#pragma unroll
    for (int nt = 0; nt < 4; ++nt) {
        const int n0 = nt * 16;
        v8f acc = {};
#pragma unroll
        for (int c = 0; c < 6; ++c) {
            const int j0 = srow + 4 * c + 2 * half;     // staged-x row for this lane's K pair
            v2f bfrag;
            bfrag[0] = xt[j0 * kRSTR + n0 + mrow];
            bfrag[1] = xt[(j0 + 1) * kRSTR + n0 + mrow];
            // D = A * B + C  (fp32, 16x16x4)
            acc = __builtin_amdgcn_wmma_f32_16x16x4_f32(
                false, afrag[c], false, bfrag, (short)0, acc, false, false);
        }
        // C/D layout: VGPR r -> (M = r + 8*half, N = mrow)
#pragma unroll
        for (int r = 0; r < 8; ++r) {
            const int mo = r + 8 * half;
            oBase[(size_t)mo * kD + n0 + mrow] = acc[r];
        }
    }
}

extern "C" void kernel_launch(void* const* d_in, const int* in_sizes, int n_in,
                              void* d_out, int out_size, void* d_ws, size_t ws_size,
                              hipStream_t stream) {
    const float* x    = (const float*)d_in[0];   // [8, 4096, 768] fp32
    const float* filt = (const float*)d_in[1];   // [8, 4096, 108] fp32
    float* out        = (float*)d_out;           // [8, 4096, 12, 64] fp32

    const int grid = kB * kH * (kS / kSBLK);     // 3072 workgroups of 256 threads
    ConvBertLightConv_wmma_kernel<<<grid, 256, 0, stream>>>(x, filt, out);
}